// GATNet_89094801588995
// MI455X (gfx1250) — compile-verified
//
#include <hip/hip_runtime.h>
#include <hip/hip_bf16.h>

typedef float v2f __attribute__((ext_vector_type(2)));
typedef float v8f __attribute__((ext_vector_type(8)));

#define NEG_SLOPE 0.2f
#define INF_EPS   1e-16f

// ---------- monotone float <-> uint encoding for atomicMax on floats ----------
__device__ __forceinline__ unsigned fEnc(float x) {
    unsigned u = __float_as_uint(x);
    return (u & 0x80000000u) ? ~u : (u | 0x80000000u);
}
__device__ __forceinline__ float fDec(unsigned u) {
    u = (u & 0x80000000u) ? (u & 0x7FFFFFFFu) : ~u;
    return __uint_as_float(u);
}
__device__ __forceinline__ float leaky(float x) { return x > 0.f ? x : NEG_SLOPE * x; }

// =====================================================================
// fp32 WMMA GEMM: C[M,NCOLS] = A[M,K] @ B[K,NCOLS]   (K, NCOLS compile-time)
// one wave per 16x16 output tile; grid.x = M/16, block = 32*(NCOLS/16)
// A-frag (16x4): lane holds M=lane&15, K=2*(lane>>4)+v  (v = vector elem)
// B-frag (4x16): lane holds N=lane&15, K=2*(lane>>4)+v
// C/D  (16x16): lane holds N=lane&15, M=8*(lane>>4)+r   (r = vector elem)
// Fully unrolled K loop -> loads use immediate offsets, pure load/WMMA stream.
// =====================================================================
template <int K, int NCOLS>
__global__ __launch_bounds__(256) void gemm_wmma_f32(
    const float* __restrict__ A, const float* __restrict__ B,
    float* __restrict__ C)
{
    const int lane = threadIdx.x & 31;
    const int wave = threadIdx.x >> 5;
    const int row  = blockIdx.x * 16 + (lane & 15);   // A row
    const int colB = wave * 16 + (lane & 15);         // B/C col
    const int kh   = (lane >> 4) << 1;                // 0 or 2

    const float* __restrict__ aPtr = A + (size_t)row * K + kh;
    const float* __restrict__ bPtr = B + (size_t)kh * NCOLS + colB;

    v8f acc = {};
#pragma unroll
    for (int k0 = 0; k0 < K; k0 += 4) {
        v2f a, b;
        a.x = aPtr[k0];
        a.y = aPtr[k0 + 1];
        b.x = bPtr[k0 * NCOLS];
        b.y = bPtr[k0 * NCOLS + NCOLS];
        acc = __builtin_amdgcn_wmma_f32_16x16x4_f32(
            false, a, false, b, (short)0, acc, false, false);
    }
    float* __restrict__ cPtr =
        C + ((size_t)blockIdx.x * 16 + ((lane >> 4) << 3)) * NCOLS + colB;
#pragma unroll
    for (int r = 0; r < 8; ++r)
        cPtr[r * NCOLS] = acc[r];
}

// ---------- per-node attention logits, layer 1 (H=4, C=32) ----------
__global__ void attn_logits_h4(const float* __restrict__ h,
                               const float* __restrict__ attS,
                               const float* __restrict__ attD,
                               float* __restrict__ alS, float* __restrict__ alD, int N)
{
    int t = blockIdx.x * blockDim.x + threadIdx.x;
    if (t >= N * 4) return;
    int n = t >> 2, hd = t & 3;
    const float* hp = h + (size_t)n * 128 + hd * 32;
    const float* as = attS + hd * 32;
    const float* ad = attD + hd * 32;
    float s = 0.f, d = 0.f;
#pragma unroll
    for (int c = 0; c < 32; ++c) { float v = hp[c]; s += v * as[c]; d += v * ad[c]; }
    alS[t] = s; alD[t] = d;
}

// ---------- per-node attention logits, layer 2 (H=1, C=64) ----------
__global__ void attn_logits_h1(const float* __restrict__ h,
                               const float* __restrict__ attS,
                               const float* __restrict__ attD,
                               float* __restrict__ alS, float* __restrict__ alD, int N)
{
    int n = blockIdx.x * blockDim.x + threadIdx.x;
    if (n >= N) return;
    const float* hp = h + (size_t)n * 64;
    float s = 0.f, d = 0.f;
#pragma unroll
    for (int c = 0; c < 64; ++c) { float v = hp[c]; s += v * attS[c]; d += v * attD[c]; }
    alS[n] = s; alD[n] = d;
}

// ---------- edge passes, layer 1 (4 heads, vectorized) ----------
__global__ void edge_max_h4(const int* __restrict__ esrc, const int* __restrict__ edst,
                            const float4* __restrict__ alS, const float4* __restrict__ alD,
                            unsigned* __restrict__ mEnc, int E, int Etot)
{
    int e = blockIdx.x * blockDim.x + threadIdx.x;
    if (e >= Etot) return;
    int s, d;
    if (e < E) { s = esrc[e]; d = edst[e]; } else { s = d = e - E; }
    float4 vs = alS[s], vd = alD[d];
    float ev[4] = { leaky(vs.x + vd.x), leaky(vs.y + vd.y),
                    leaky(vs.z + vd.z), leaky(vs.w + vd.w) };
#pragma unroll
    for (int h = 0; h < 4; ++h) atomicMax(&mEnc[d * 4 + h], fEnc(ev[h]));
}

__global__ void edge_exp_h4(const int* __restrict__ esrc, const int* __restrict__ edst,
                            const float4* __restrict__ alS, const float4* __restrict__ alD,
                            const unsigned* __restrict__ mEnc,
                            float* __restrict__ ex, float* __restrict__ den, int E, int Etot)
{
    int e = blockIdx.x * blockDim.x + threadIdx.x;
    if (e >= Etot) return;
    int s, d;
    if (e < E) { s = esrc[e]; d = edst[e]; } else { s = d = e - E; }
    float4 vs = alS[s], vd = alD[d];
    float ev[4] = { leaky(vs.x + vd.x), leaky(vs.y + vd.y),
                    leaky(vs.z + vd.z), leaky(vs.w + vd.w) };
#pragma unroll
    for (int h = 0; h < 4; ++h) {
        float x = __expf(ev[h] - fDec(mEnc[d * 4 + h]));
        ex[(size_t)e * 4 + h] = x;
        atomicAdd(&den[d * 4 + h], x);
    }
}

// one thread per (edge, feature c in 0..127); head = c>>5
__global__ void edge_agg_128(const int* __restrict__ esrc, const int* __restrict__ edst,
                             const float* __restrict__ ex, const float* __restrict__ den,
                             const float* __restrict__ h, float* __restrict__ out,
                             int E, int Etot)
{
    int t = blockIdx.x * blockDim.x + threadIdx.x;
    if (t >= Etot * 128) return;
    int e = t >> 7, c = t & 127, hd = c >> 5;
    int s, d;
    if (e < E) { s = esrc[e]; d = edst[e]; } else { s = d = e - E; }
    float alpha = ex[(size_t)e * 4 + hd] / (den[d * 4 + hd] + INF_EPS);
    atomicAdd(&out[(size_t)d * 128 + c], alpha * h[(size_t)s * 128 + c]);
}

__global__ void finalize_elu(float* __restrict__ a, const float* __restrict__ b, int N)
{
    int t = blockIdx.x * blockDim.x + threadIdx.x;
    if (t >= N * 128) return;
    float v = a[t] + b[t & 127];
    a[t] = v > 0.f ? v : __expf(v) - 1.f;
}

// ---------- edge passes, layer 2 (1 head) ----------
__global__ void edge_max_h1(const int* __restrict__ esrc, const int* __restrict__ edst,
                            const float* __restrict__ alS, const float* __restrict__ alD,
                            unsigned* __restrict__ mEnc, int E, int Etot)
{
    int e = blockIdx.x * blockDim.x + threadIdx.x;
    if (e >= Etot) return;
    int s, d;
    if (e < E) { s = esrc[e]; d = edst[e]; } else { s = d = e - E; }
    atomicMax(&mEnc[d], fEnc(leaky(alS[s] + alD[d])));
}

__global__ void edge_exp_h1(const int* __restrict__ esrc, const int* __restrict__ edst,
                            const float* __restrict__ alS, const float* __restrict__ alD,
                            const unsigned* __restrict__ mEnc,
                            float* __restrict__ ex, float* __restrict__ den, int E, int Etot)
{
    int e = blockIdx.x * blockDim.x + threadIdx.x;
    if (e >= Etot) return;
    int s, d;
    if (e < E) { s = esrc[e]; d = edst[e]; } else { s = d = e - E; }
    float x = __expf(leaky(alS[s] + alD[d]) - fDec(mEnc[d]));
    ex[e] = x;
    atomicAdd(&den[d], x);
}

__global__ void edge_agg_64(const int* __restrict__ esrc, const int* __restrict__ edst,
                            const float* __restrict__ ex, const float* __restrict__ den,
                            const float* __restrict__ h, float* __restrict__ out,
                            int E, int Etot)
{
    int t = blockIdx.x * blockDim.x + threadIdx.x;
    if (t >= Etot * 64) return;
    int e = t >> 6, c = t & 63;
    int s, d;
    if (e < E) { s = esrc[e]; d = edst[e]; } else { s = d = e - E; }
    float alpha = ex[e] / (den[d] + INF_EPS);
    atomicAdd(&out[(size_t)d * 64 + c], alpha * h[(size_t)s * 64 + c]);
}

__global__ void finalize_bias64(float* __restrict__ o, const float* __restrict__ b, int N)
{
    int t = blockIdx.x * blockDim.x + threadIdx.x;
    if (t >= N * 64) return;
    o[t] += b[t & 63];
}

// =====================================================================
extern "C" void kernel_launch(void* const* d_in, const int* in_sizes, int n_in,
                              void* d_out, int out_size, void* d_ws, size_t ws_size,
                              hipStream_t stream)
{
    const float* x    = (const float*)d_in[0];
    const int*   eidx = (const int*)d_in[1];
    const float* W1   = (const float*)d_in[2];
    const float* aS1  = (const float*)d_in[3];
    const float* aD1  = (const float*)d_in[4];
    const float* b1   = (const float*)d_in[5];
    const float* W2   = (const float*)d_in[6];
    const float* aS2  = (const float*)d_in[7];
    const float* aD2  = (const float*)d_in[8];
    const float* b2   = (const float*)d_in[9];
    float* out = (float*)d_out;

    const int N    = in_sizes[0] / 128;     // 50000
    const int E    = in_sizes[1] / 2;       // 800000
    const int Etot = E + N;                 // + self loops
    const int* esrc = eidx;
    const int* edst = eidx + E;

    // -------- workspace layout (floats) --------
    float* ws = (float*)d_ws;
    size_t o = 0;
    float*    h1   = ws + o; o += (size_t)N * 128;
    float*    alS1 = ws + o; o += (size_t)N * 4;
    float*    alD1 = ws + o; o += (size_t)N * 4;
    float*    ex1  = ws + o; o += (size_t)Etot * 4;
    float*    agg1 = ws + o; o += (size_t)N * 128;
    unsigned* m1   = (unsigned*)(ws + o); o += (size_t)N * 4;
    float*    den1 = ws + o; o += (size_t)N * 4;
    float*    h2   = ws + o; o += (size_t)N * 64;
    float*    alS2 = ws + o; o += (size_t)N;
    float*    alD2 = ws + o; o += (size_t)N;
    float*    ex2  = ws + o; o += (size_t)Etot;
    unsigned* m2   = (unsigned*)(ws + o); o += (size_t)N;
    float*    den2 = ws + o; o += (size_t)N;

    const int TB = 256;
    auto cdiv = [](int a, int b) { return (a + b - 1) / b; };

    // -------- zero the reduction buffers (graph-capture-safe memsets) --------
    hipMemsetAsync(m1,   0, (size_t)N * 4 * sizeof(unsigned), stream);
    hipMemsetAsync(den1, 0, (size_t)N * 4 * sizeof(float), stream);
    hipMemsetAsync(agg1, 0, (size_t)N * 128 * sizeof(float), stream);
    hipMemsetAsync(m2,   0, (size_t)N * sizeof(unsigned), stream);
    hipMemsetAsync(den2, 0, (size_t)N * sizeof(float), stream);
    hipMemsetAsync(out,  0, (size_t)N * 64 * sizeof(float), stream);

    // -------- layer 1 --------
    gemm_wmma_f32<128, 128><<<N / 16, 256, 0, stream>>>(x, W1, h1);
    attn_logits_h4<<<cdiv(N * 4, TB), TB, 0, stream>>>(h1, aS1, aD1, alS1, alD1, N);
    edge_max_h4<<<cdiv(Etot, TB), TB, 0, stream>>>(esrc, edst,
        (const float4*)alS1, (const float4*)alD1, m1, E, Etot);
    edge_exp_h4<<<cdiv(Etot, TB), TB, 0, stream>>>(esrc, edst,
        (const float4*)alS1, (const float4*)alD1, m1, ex1, den1, E, Etot);
    edge_agg_128<<<cdiv(Etot * 128, TB), TB, 0, stream>>>(esrc, edst,
        ex1, den1, h1, agg1, E, Etot);
    finalize_elu<<<cdiv(N * 128, TB), TB, 0, stream>>>(agg1, b1, N);

    // -------- layer 2 --------
    gemm_wmma_f32<128, 64><<<N / 16, 128, 0, stream>>>(agg1, W2, h2);
    attn_logits_h1<<<cdiv(N, TB), TB, 0, stream>>>(h2, aS2, aD2, alS2, alD2, N);
    edge_max_h1<<<cdiv(Etot, TB), TB, 0, stream>>>(esrc, edst, alS2, alD2, m2, E, Etot);
    edge_exp_h1<<<cdiv(Etot, TB), TB, 0, stream>>>(esrc, edst, alS2, alD2, m2, ex2, den2, E, Etot);
    edge_agg_64<<<cdiv(Etot * 64, TB), TB, 0, stream>>>(esrc, edst, ex2, den2, h2, out, E, Etot);
    finalize_bias64<<<cdiv(N * 64, TB), TB, 0, stream>>>(out, b2, N);

    (void)n_in; (void)out_size; (void)ws_size;
}